// RX_73813307949712
// MI455X (gfx1250) — compile-verified
//
#include <hip/hip_runtime.h>

typedef __attribute__((ext_vector_type(2))) float v2f;
typedef __attribute__((ext_vector_type(8))) float v8f;

#define NSTATE 4096                // 2^12
#define NBATCH 256
#define PLANE  (NSTATE * NBATCH)   // floats per (re or im) plane

// ---------------------------------------------------------------------------
// Build the three 16x16 complex group matrices G = kron(M_{4g},..,M_{4g+3}).
// M[j][k]: diag = cos(th/2) (real), offdiag = -i sin(th/2) (imaginary).
// Layout in G: for group g: Gr at [g*512 .. +256), Gi at [g*512+256 .. +512).
// Row-major 16x16: G[r*16 + c].
// ---------------------------------------------------------------------------
__global__ __launch_bounds__(256) void rx_gate_prep(const float* __restrict__ angle,
                                                    float* __restrict__ G) {
    int e = threadIdx.x;          // 0..255 -> (r, c)
    int r = e >> 4;
    int c = e & 15;
    for (int g = 0; g < 3; ++g) {
        float pr = 1.0f, pi = 0.0f;
        for (int w = 0; w < 4; ++w) {
            float th = 0.5f * angle[4 * g + w];
            int rb = (r >> (3 - w)) & 1;
            int cb = (c >> (3 - w)) & 1;
            float mr, mi;
            if (rb == cb) { mr = __cosf(th); mi = 0.0f; }
            else          { mr = 0.0f;      mi = -__sinf(th); }
            float nr = pr * mr - pi * mi;
            float ni = pr * mi + pi * mr;
            pr = nr; pi = ni;
        }
        G[g * 512 + e]       = pr;
        G[g * 512 + 256 + e] = pi;
    }
}

// ---------------------------------------------------------------------------
// One contraction pass: for every 16x16 tile (rows = contracted group index,
// cols = 16 batch columns), D = G * S  (complex), via chained
// V_WMMA_F32_16X16X4_F32 (4 chunks of K=4 -> 16x16x16, exact f32).
//
// MODE 0: input real-only (x), outputs SoA planes (outR/outI)
// MODE 1: input SoA planes, outputs SoA planes
// MODE 2: input SoA planes, output interleaved complex64 into outR
//
// Flattened element index of (n, t) is n*256 + t.  uStrideN/vStrideN are the
// n-strides of the two non-contracted group indices; rowStrideF is the float
// (=element) stride of the contracted index.
// ---------------------------------------------------------------------------
template <int MODE>
__global__ __launch_bounds__(256) void rx_pass(
    const float* __restrict__ inR, const float* __restrict__ inI,
    float* __restrict__ outR, float* __restrict__ outI,
    const float* __restrict__ Gr, const float* __restrict__ Gi,
    int uStrideN, int vStrideN, int rowStrideF)
{
    const int lane = threadIdx.x & 31;
    const int wave = threadIdx.x >> 5;
    const int tileIdx = blockIdx.x * 8 + wave;   // 512 blocks * 8 waves = 4096 tiles

    // ---- A fragment (gate matrix), 16x4 f32 per chunk: 2 VGPRs ----
    // lane = m + 16*(kpair), component x -> K = 2*kpair + 4j, y -> +1
    const int m  = lane & 15;
    const int kb = (lane >> 4) << 1;
    v2f aR[4], aI[4], aIn[4];
#pragma unroll
    for (int j = 0; j < 4; ++j) {
        int col = 4 * j + kb;
        aR[j].x  = Gr[m * 16 + col];
        aR[j].y  = Gr[m * 16 + col + 1];
        aI[j].x  = Gi[m * 16 + col];
        aI[j].y  = Gi[m * 16 + col + 1];
        aIn[j].x = -aI[j].x;
        aIn[j].y = -aI[j].y;
    }

    // tile coordinates: tt = batch tile, (u, v) = non-contracted group indices
    const int tt = tileIdx & 15;
    const int v  = (tileIdx >> 4) & 15;
    const int u  = tileIdx >> 8;
    const int base = (u * uStrideN + v * vStrideN) * NBATCH + tt * 16;

    // ---- B fragments (state tile), 4x16 f32 per chunk: 2 VGPRs ----
    // lane = n + 16*g, component x -> row K = 4j + 2g, y -> 4j + 2g + 1
    const int nCol = lane & 15;
    const int gSel = lane >> 4;
    v2f bR[4], bI[4];
#pragma unroll
    for (int j = 0; j < 4; ++j) {
        int r0 = base + (4 * j + 2 * gSel) * rowStrideF + nCol;
        bR[j].x = inR[r0];
        bR[j].y = inR[r0 + rowStrideF];
        if (MODE != 0) {
            bI[j].x = inI[r0];
            bI[j].y = inI[r0 + rowStrideF];
        }
    }

    // ---- complex GEMM: Dr = Gr*Sr - Gi*Si ; Di = Gr*Si + Gi*Sr ----
    v8f dR = {};
    v8f dI = {};
#pragma unroll
    for (int j = 0; j < 4; ++j)
        dR = __builtin_amdgcn_wmma_f32_16x16x4_f32(false, aR[j], false, bR[j],
                                                   (short)0, dR, false, false);
#pragma unroll
    for (int j = 0; j < 4; ++j)
        dI = __builtin_amdgcn_wmma_f32_16x16x4_f32(false, aI[j], false, bR[j],
                                                   (short)0, dI, false, false);
    if (MODE != 0) {
#pragma unroll
        for (int j = 0; j < 4; ++j)
            dR = __builtin_amdgcn_wmma_f32_16x16x4_f32(false, aIn[j], false, bI[j],
                                                       (short)0, dR, false, false);
#pragma unroll
        for (int j = 0; j < 4; ++j)
            dI = __builtin_amdgcn_wmma_f32_16x16x4_f32(false, aR[j], false, bI[j],
                                                       (short)0, dI, false, false);
    }

    // ---- store D (C/D layout: VGPR r holds row r + 8*(lane>=16), col = lane&15)
    if (MODE == 2) {
#pragma unroll
        for (int rr = 0; rr < 8; ++rr) {
            int e = base + (rr + 8 * gSel) * rowStrideF + nCol;
            v2f val; val.x = dR[rr]; val.y = dI[rr];
            *(v2f*)(outR + 2 * e) = val;       // interleaved complex64
        }
    } else {
#pragma unroll
        for (int rr = 0; rr < 8; ++rr) {
            int e = base + (rr + 8 * gSel) * rowStrideF + nCol;
            outR[e] = dR[rr];
            outI[e] = dI[rr];
        }
    }
}

extern "C" void kernel_launch(void* const* d_in, const int* in_sizes, int n_in,
                              void* d_out, int out_size, void* d_ws, size_t ws_size,
                              hipStream_t stream) {
    const float* x     = (const float*)d_in[0];   // [4096, 256] f32
    const float* angle = (const float*)d_in[1];   // [12] f32
    float* out = (float*)d_out;                   // 4096*256 complex64 = 2M floats
    float* ws  = (float*)d_ws;

    // workspace layout: [0,PLANE) SR, [PLANE,2*PLANE) SI, then 3*512 floats of G
    float* SR = ws;
    float* SI = ws + PLANE;
    float* G  = ws + 2 * PLANE;
    // d_out doubles as SoA scratch for the pass-1 result
    float* DR = out;
    float* DI = out + PLANE;

    rx_gate_prep<<<1, 256, 0, stream>>>(angle, G);

    const float* GrA = G + 0 * 512, *GiA = G + 0 * 512 + 256;  // wires 0-3  (n>>8)
    const float* GrB = G + 1 * 512, *GiB = G + 1 * 512 + 256;  // wires 4-7  ((n>>4)&15)
    const float* GrC = G + 2 * 512, *GiC = G + 2 * 512 + 256;  // wires 8-11 (n&15)

    // Pass 1: contract group C (n-stride 1): real x -> d_out SoA
    rx_pass<0><<<512, 256, 0, stream>>>(x, nullptr, DR, DI, GrC, GiC,
                                        /*u=a*/256, /*v=b*/16, /*row*/256);
    // Pass 2: contract group B (n-stride 16): d_out SoA -> ws SoA
    rx_pass<1><<<512, 256, 0, stream>>>(DR, DI, SR, SI, GrB, GiB,
                                        /*u=a*/256, /*v=c*/1, /*row*/4096);
    // Pass 3: contract group A (n-stride 256): ws SoA -> d_out interleaved complex64
    rx_pass<2><<<512, 256, 0, stream>>>(SR, SI, out, nullptr, GrA, GiA,
                                        /*u=b*/16, /*v=c*/1, /*row*/65536);
}